// GraphPooling_5239860101878
// MI455X (gfx1250) — compile-verified
//
#include <hip/hip_runtime.h>
#include <stdint.h>

// Problem constants (match reference: B=8, N=3072, D=1024, POOL=3)
namespace {
constexpr int kB     = 8;
constexpr int kN     = 3072;
constexpr int kD     = 1024;
constexpr int kPool  = 3;
constexpr int kSteps = kN / kPool;   // 1024
constexpr float kEps = 0.006f;
}

typedef __attribute__((ext_vector_type(2))) float v2f;
typedef __attribute__((ext_vector_type(8))) float v8f;

// ---- CDNA5 async global->LDS copy (tracked by ASYNCcnt) -------------------
// Flat address of a __shared__ object: low 32 bits are the LDS byte offset
// (LDS aperture rule).
__device__ __forceinline__ unsigned lds_u32(const void* p) {
  return (unsigned)(uintptr_t)p;
}
__device__ __forceinline__ void async_copy_b128(unsigned lds_byte_addr,
                                                unsigned long long gaddr) {
  asm volatile("global_load_async_to_lds_b128 %0, %1, off"
               :: "v"(lds_byte_addr), "v"(gaddr) : "memory");
}
__device__ __forceinline__ void wait_async_zero() {
  asm volatile("s_wait_asynccnt 0" ::: "memory");
}

// ---------------------------------------------------------------------------
// Kernel 1: stage 3 rows of x into LDS via async copies, emit
//   W[b,k,d]   = x[b,3k]+x[b,3k+1]+x[b,3k+2]   (chunk sums, to workspace)
//   out[b,2k+1]= W/3                           (window means)
// ---------------------------------------------------------------------------
__global__ __launch_bounds__(256) void win_pool_kernel(
    const float* __restrict__ x, float* __restrict__ wsum,
    float* __restrict__ out) {
  __shared__ float rows[kPool * kD];          // 12 KB
  const int bk  = blockIdx.x;
  const int b   = bk / kSteps;
  const int k   = bk % kSteps;
  const int tid = threadIdx.x;

  const float* src = x + ((size_t)b * kN + (size_t)kPool * k) * kD;
#pragma unroll
  for (int r = 0; r < kPool; ++r) {
    // each lane moves 16B; 256 threads x 3 rows covers 12 KB
    unsigned lds = lds_u32(&rows[r * kD + tid * 4]);
    async_copy_b128(lds,
        (unsigned long long)(uintptr_t)(src + (size_t)r * kD + tid * 4));
  }
  wait_async_zero();
  __syncthreads();

#pragma unroll
  for (int i = 0; i < kD / 256; ++i) {
    const int d = tid + 256 * i;
    const float w = rows[d] + rows[kD + d] + rows[2 * kD + d];
    wsum[((size_t)b * kSteps + k) * kD + d] = w;
    out[((size_t)b * 2 * kSteps + 2 * k + 1) * kD + d] = w * (1.0f / 3.0f);
  }
}

// ---------------------------------------------------------------------------
// Kernel 2: in-place inclusive scan of W along k, per (b, 16-wide d tile).
// Scan-as-matmul: P_tile(16x16) = T(16x16, lower-triangular ones) * W_tile,
// done as four accumulating V_WMMA_F32_16X16X4_F32, + per-column carry.
// One wave32 per block (WMMA requires EXEC all ones).
// ---------------------------------------------------------------------------
__global__ __launch_bounds__(32) void scan_wmma_kernel(float* __restrict__ P) {
  const int lane = threadIdx.x;               // 0..31
  const int b    = blockIdx.x / (kD / 16);
  const int d0   = (blockIdx.x % (kD / 16)) * 16;
  const int n    = lane & 15;                 // column within tile
  const int hi   = lane >> 4;                 // half-wave select

  float carry = 0.0f;                         // running prefix per column

  for (int k0 = 0; k0 < kSteps; k0 += 16) {
    v8f c = {};
#pragma unroll
    for (int t = 0; t < 4; ++t) {
      // A tile (16x4 f32 layout): lanes 0-15 hold K={4t+0,4t+1},
      // lanes 16-31 hold K={4t+2,4t+3}; row m = lane&15.
      // T[m,k] = 1 iff k <= m  (inclusive scan).
      const int j0 = 4 * t + 2 * hi;
      v2f a;
      a.x = (j0     <= n) ? 1.0f : 0.0f;
      a.y = (j0 + 1 <= n) ? 1.0f : 0.0f;
      // B tile (4x16 f32): row K striped across half-waves, col = lane&15.
      const int r0 = k0 + 4 * t + 2 * hi;
      v2f bm;
      bm.x = P[((size_t)b * kSteps + r0    ) * kD + d0 + n];
      bm.y = P[((size_t)b * kSteps + r0 + 1) * kD + d0 + n];
      c = __builtin_amdgcn_wmma_f32_16x16x4_f32(
          /*neg_a=*/false, a, /*neg_b=*/false, bm,
          /*c_mod=*/(short)0, c, /*reuse_a=*/false, /*reuse_b=*/false);
    }
    // Add carry, write inclusive prefix back in place.
    // C layout: VGPR v holds row v (lanes 0-15) and row v+8 (lanes 16-31).
#pragma unroll
    for (int v = 0; v < 8; ++v) {
      const float val = c[v] + carry;
      const int row = v + 8 * hi;
      P[((size_t)b * kSteps + k0 + row) * kD + d0 + n] = val;
      c[v] = val;
    }
    // New carry = row 15 (c[7] of upper-half lanes), broadcast per column.
    carry = __shfl(c[7], 16 + n, 32);
  }
}

// ---------------------------------------------------------------------------
// Kernel 3: segment means from coarse prefix P plus <=2 partial x rows/edge.
//   c[m] = (m/3 > 0 ? P[b, m/3 - 1, :] : 0) + sum_{i < m%3} x[b, 3*(m/3)+i, :]
//   out[b,2k] = (c[e+1] - c[s]) / (e-s+1) + EPS
// ---------------------------------------------------------------------------
__global__ __launch_bounds__(256) void seg_mean_kernel(
    const float* __restrict__ x, const long long* __restrict__ graph,
    const float* __restrict__ P, float* __restrict__ out) {
  const int bk  = blockIdx.x;
  const int b   = bk / kSteps;
  const int k   = bk % kSteps;
  const int d   = threadIdx.x * 4;            // float4 per thread

  const long long s = graph[2 * (size_t)bk + 0];
  const long long e = graph[2 * (size_t)bk + 1];
  const float inv_len = 1.0f / (float)(e - s + 1);

  auto cval = [&](long long m) -> float4 {
    const long long q = m / 3;
    const int rem = (int)(m - 3 * q);
    float4 v = make_float4(0.f, 0.f, 0.f, 0.f);
    if (q > 0) {
      v = *(const float4*)&P[((size_t)b * kSteps + (size_t)(q - 1)) * kD + d];
    }
    for (int i = 0; i < rem; ++i) {
      const float4 t =
          *(const float4*)&x[((size_t)b * kN + (size_t)(3 * q + i)) * kD + d];
      v.x += t.x; v.y += t.y; v.z += t.z; v.w += t.w;
    }
    return v;
  };

  const float4 ce = cval(e + 1);
  const float4 cs = cval(s);
  float4 r;
  r.x = (ce.x - cs.x) * inv_len + kEps;
  r.y = (ce.y - cs.y) * inv_len + kEps;
  r.z = (ce.z - cs.z) * inv_len + kEps;
  r.w = (ce.w - cs.w) * inv_len + kEps;
  *(float4*)&out[((size_t)b * 2 * kSteps + 2 * (size_t)k) * kD + d] = r;
}

// ---------------------------------------------------------------------------
extern "C" void kernel_launch(void* const* d_in, const int* in_sizes, int n_in,
                              void* d_out, int out_size, void* d_ws,
                              size_t ws_size, hipStream_t stream) {
  (void)in_sizes; (void)n_in; (void)out_size; (void)ws_size;
  const float*      x     = (const float*)d_in[0];
  const long long*  graph = (const long long*)d_in[1];   // int64 [B,steps,2]
  float*            out   = (float*)d_out;               // [B, 2*steps, D]
  float*            W     = (float*)d_ws;                // B*steps*D f32 = 32 MB

  win_pool_kernel<<<kB * kSteps, 256, 0, stream>>>(x, W, out);
  scan_wmma_kernel<<<kB * (kD / 16), 32, 0, stream>>>(W);
  seg_mean_kernel<<<kB * kSteps, 256, 0, stream>>>(x, graph, W, out);
}